// ConcatAttention_79894981640368
// MI455X (gfx1250) — compile-verified
//
#include <hip/hip_runtime.h>
#include <hip/hip_bf16.h>

// Problem constants (reference: T, S, B, H = 32, 128, 16, 1024)
constexpr int Tn = 32;
constexpr int Sn = 128;
constexpr int Bn = 16;
constexpr int Hn = 1024;

typedef __attribute__((ext_vector_type(2))) float v2f;
typedef __attribute__((ext_vector_type(8))) float v8f;

// ---------------------------------------------------------------------------
// tanh: gfx1250 has a native V_TANH_F32 transcendental (tanh-insts feature).
// Fall back to exp2-based formula if the builtin is missing.
// ---------------------------------------------------------------------------
__device__ __forceinline__ float fast_tanh(float x) {
#if __has_builtin(__builtin_amdgcn_tanhf)
  return __builtin_amdgcn_tanhf(x);
#else
  // tanh(x) = 1 - 2/(e^{2x}+1);  e^{2x} = 2^{x * 2*log2(e)}
  float e = __builtin_amdgcn_exp2f(x * 2.8853900817779268f);
  return 1.0f - 2.0f * __builtin_amdgcn_rcpf(e + 1.0f);
#endif
}

// ---------------------------------------------------------------------------
// Kernel 1: f32 GEMM via V_WMMA_F32_16X16X4_F32.
//   out[row][o] = sum_k A[row][k] * W[o][colOff + k]   (+ bias[o] if given)
// A: rows x Hn row-major.  W: Hn x 2*Hn row-major (Wa_w).  out: rows x Hn.
// One wave computes a 16x64 tile (4 accumulators share the A fragment).
// Fragment layouts per ISA 7.12.2 (32-bit A 16x4 / B 4x16):
//   lane = half*16 + l16;  VGPR r holds K = half*2 + r  -> contiguous float2.
// ---------------------------------------------------------------------------
__global__ void proj16x16x4(const float* __restrict__ A,
                            const float* __restrict__ W,
                            const float* __restrict__ bias,
                            float* __restrict__ out,
                            int colOff) {
  const int lane = threadIdx.x & 31;
  const int wave = threadIdx.x >> 5;
  const int l16  = lane & 15;
  const int half = lane >> 4;          // 0 or 1
  const int kSub = half * 2;           // K offset of this lane's pair
  const int rowBase = (blockIdx.y * 4 + wave) * 16;
  const int colBase = blockIdx.x * 64;

  const float* aP = A + (size_t)(rowBase + l16) * Hn + kSub;
  const float* b0 = W + (size_t)(colBase +  0 + l16) * (2 * Hn) + colOff + kSub;
  const float* b1 = W + (size_t)(colBase + 16 + l16) * (2 * Hn) + colOff + kSub;
  const float* b2 = W + (size_t)(colBase + 32 + l16) * (2 * Hn) + colOff + kSub;
  const float* b3 = W + (size_t)(colBase + 48 + l16) * (2 * Hn) + colOff + kSub;

  v8f acc0 = {}, acc1 = {}, acc2 = {}, acc3 = {};

#pragma unroll 4
  for (int k = 0; k < Hn; k += 4) {
    v2f a  = *(const v2f*)(aP + k);
    v2f w0 = *(const v2f*)(b0 + k);
    v2f w1 = *(const v2f*)(b1 + k);
    v2f w2 = *(const v2f*)(b2 + k);
    v2f w3 = *(const v2f*)(b3 + k);
    acc0 = __builtin_amdgcn_wmma_f32_16x16x4_f32(false, a, false, w0, (short)0, acc0, false, false);
    acc1 = __builtin_amdgcn_wmma_f32_16x16x4_f32(false, a, false, w1, (short)0, acc1, false, false);
    acc2 = __builtin_amdgcn_wmma_f32_16x16x4_f32(false, a, false, w2, (short)0, acc2, false, false);
    acc3 = __builtin_amdgcn_wmma_f32_16x16x4_f32(false, a, false, w3, (short)0, acc3, false, false);
  }

  float bv0 = 0.f, bv1 = 0.f, bv2 = 0.f, bv3 = 0.f;
  if (bias) {
    bv0 = bias[colBase +  0 + l16];
    bv1 = bias[colBase + 16 + l16];
    bv2 = bias[colBase + 32 + l16];
    bv3 = bias[colBase + 48 + l16];
  }

  // C/D layout: VGPR r, lanes 0-15 -> M=r ; lanes 16-31 -> M=r+8 ; N = l16
  const int rowW = rowBase + half * 8;
#pragma unroll
  for (int r = 0; r < 8; ++r) {
    const size_t row = (size_t)(rowW + r);
    out[row * Hn + colBase +  0 + l16] = acc0[r] + bv0;
    out[row * Hn + colBase + 16 + l16] = acc1[r] + bv1;
    out[row * Hn + colBase + 32 + l16] = acc2[r] + bv2;
    out[row * Hn + colBase + 48 + l16] = acc3[r] + bv3;
  }
}

// ---------------------------------------------------------------------------
// Kernel 2: scores[t,s,b] = sum_h Va[h] * tanh(htp[t,b,h] + srcp[s,b,h])
// (bias already folded into htp). One block per (t,b); htp row + Va in LDS;
// 8 waves, each wave sweeps 16 s-values and does a wave-shuffle reduction.
// ---------------------------------------------------------------------------
__global__ void scores_kernel(const float* __restrict__ htp,
                              const float* __restrict__ srcp,
                              const float* __restrict__ Va,
                              float* __restrict__ out) {
  __shared__ float sx[Hn];
  __shared__ float sv[Hn];

  const int b = blockIdx.x;
  const int t = blockIdx.y;
  const int lane = threadIdx.x & 31;
  const int wave = threadIdx.x >> 5;

  const float* xt = htp + (size_t)(t * Bn + b) * Hn;
  for (int h = threadIdx.x * 4; h < Hn; h += blockDim.x * 4) {
    *(float4*)&sx[h] = *(const float4*)&xt[h];
    *(float4*)&sv[h] = *(const float4*)&Va[h];
  }
  __syncthreads();

  for (int si = 0; si < Sn / 8; ++si) {
    const int s = wave + si * 8;
    const float* y = srcp + (size_t)(s * Bn + b) * Hn;
    float acc = 0.f;
#pragma unroll
    for (int i = 0; i < 8; ++i) {
      const int h = i * 128 + lane * 4;
      float4 yv = *(const float4*)(y + h);
      float4 xv = *(const float4*)&sx[h];
      float4 vv = *(const float4*)&sv[h];
      acc += vv.x * fast_tanh(xv.x + yv.x);
      acc += vv.y * fast_tanh(xv.y + yv.y);
      acc += vv.z * fast_tanh(xv.z + yv.z);
      acc += vv.w * fast_tanh(xv.w + yv.w);
    }
#pragma unroll
    for (int off = 16; off > 0; off >>= 1)
      acc += __shfl_xor(acc, off, 32);
    if (lane == 0)
      out[((size_t)t * Sn + s) * Bn + b] = acc;
  }
}

// ---------------------------------------------------------------------------
// Kernel 3: in-place softmax over the S axis. One wave per (t,b) vector of
// 128 scores (4 values per lane), shuffle reductions for max and sum.
// ---------------------------------------------------------------------------
__global__ void softmax_s(float* __restrict__ out) {
  const int lane = threadIdx.x & 31;
  const int wave = threadIdx.x >> 5;
  const int wid  = blockIdx.x * 8 + wave;   // 0 .. T*B-1
  const int t = wid / Bn;
  const int b = wid % Bn;

  float v[4];
#pragma unroll
  for (int i = 0; i < 4; ++i) {
    const int s = lane + 32 * i;
    v[i] = out[((size_t)t * Sn + s) * Bn + b];
  }
  float m = fmaxf(fmaxf(v[0], v[1]), fmaxf(v[2], v[3]));
#pragma unroll
  for (int off = 16; off > 0; off >>= 1)
    m = fmaxf(m, __shfl_xor(m, off, 32));

  float e[4];
  float sum = 0.f;
#pragma unroll
  for (int i = 0; i < 4; ++i) {
    e[i] = __builtin_amdgcn_exp2f((v[i] - m) * 1.4426950408889634f);
    sum += e[i];
  }
#pragma unroll
  for (int off = 16; off > 0; off >>= 1)
    sum += __shfl_xor(sum, off, 32);

  const float inv = 1.0f / sum;
#pragma unroll
  for (int i = 0; i < 4; ++i) {
    const int s = lane + 32 * i;
    out[((size_t)t * Sn + s) * Bn + b] = e[i] * inv;
  }
}

// ---------------------------------------------------------------------------
// Launch: proj(h_t) -> proj(src) -> scores -> softmax, all on `stream`.
// Workspace: htp 512*1024 f32 (2MB) + srcp 2048*1024 f32 (8MB) in d_ws.
// ---------------------------------------------------------------------------
extern "C" void kernel_launch(void* const* d_in, const int* in_sizes, int n_in,
                              void* d_out, int out_size, void* d_ws, size_t ws_size,
                              hipStream_t stream) {
  (void)in_sizes; (void)n_in; (void)out_size; (void)ws_size;

  const float* h_t  = (const float*)d_in[0];  // (T,B,H)
  const float* src  = (const float*)d_in[1];  // (S,B,H)
  const float* Wa_w = (const float*)d_in[2];  // (H, 2H)
  const float* Wa_b = (const float*)d_in[3];  // (H,)
  const float* Va_w = (const float*)d_in[4];  // (1, H)
  float* out = (float*)d_out;                 // (T,S,B,1)

  float* htp  = (float*)d_ws;                       // (T*B, H)
  float* srcp = htp + (size_t)Tn * Bn * Hn;         // (S*B, H)

  // ht_proj = h_t · Wt^T + Wa_b   (Wt = Wa_w[:, :H])
  proj16x16x4<<<dim3(Hn / 64, (Tn * Bn) / 64), 128, 0, stream>>>(
      h_t, Wa_w, Wa_b, htp, 0);
  // src_proj = src · Ws^T          (Ws = Wa_w[:, H:])
  proj16x16x4<<<dim3(Hn / 64, (Sn * Bn) / 64), 128, 0, stream>>>(
      src, Wa_w, nullptr, srcp, Hn);

  // scores[t,s,b] = Va · tanh(htp_row + srcp_row)
  scores_kernel<<<dim3(Bn, Tn), 256, 0, stream>>>(htp, srcp, Va_w, out);

  // softmax over S, in place
  softmax_s<<<(Tn * Bn) / 8, 256, 0, stream>>>(out);
}